// Sampler_50019189129706
// MI455X (gfx1250) — compile-verified
//
#include <hip/hip_runtime.h>
#include <math.h>

// Problem constants (reference: B=128, V=128000)
#define B_ROWS   128
#define V_COLS   128000
#define SEGS     5                         // 5 segments per row -> 640 blocks
#define THREADS  256                       // 8 wave32 per block
#define VEC4_PER_SEG (V_COLS / 4 / SEGS)   // 6400 float4 per segment per array
#define STAGES   (VEC4_PER_SEG / THREADS)  // 25 pipeline stages, 1 float4/thread/stage

// ---- CDNA5 async tensor-path primitives (gfx1250) -------------------------
// GLOBAL_LOAD_ASYNC_TO_LDS_B128, GVS addressing: mem = SADDR(64) + VADDR(i32) + ioffset
// LDS dest = wave LDS base + VDST vgpr. Tracked on ASYNCcnt (in-order load completion).
__device__ __forceinline__ void async_copy_b128(unsigned lds_byte_off,
                                                unsigned g_byte_off,
                                                const float* __restrict__ sbase) {
    asm volatile("global_load_async_to_lds_b128 %0, %1, %2"
                 :: "v"(lds_byte_off), "v"(g_byte_off), "s"(sbase)
                 : "memory");
}

template <int N>
__device__ __forceinline__ void wait_asynccnt_le() {
    asm volatile("s_wait_asynccnt %0" :: "i"(N) : "memory");
}

// ---------------------------------------------------------------------------
__device__ __forceinline__ void upd(float v, int i, float& bv, int& bi) {
    // first-occurrence argmax semantics (lowest index wins ties)
    if (v > bv || (v == bv && i < bi)) { bv = v; bi = i; }
}

template <bool GREEDY>
__device__ __forceinline__ void stream_segment(const float* __restrict__ lrow,
                                               const float* __restrict__ nrow,
                                               float invt, int seg, int t,
                                               float4 (*sL)[THREADS],
                                               float4 (*sN)[THREADS],
                                               float& bv, int& bi) {
    // LDS byte offsets: generic pointer low 32 bits == wave-relative LDS address
    const unsigned baseL = (unsigned)(unsigned long long)&sL[0][0];
    const unsigned baseN = (unsigned)(unsigned long long)&sN[0][0];
    const unsigned segv  = (unsigned)(seg * VEC4_PER_SEG);

    // prologue: stage 0 -> buffer 0
    {
        unsigned g = (segv + (unsigned)t) * 16u;
        async_copy_b128(baseL + (unsigned)t * 16u, g, lrow);
        if (!GREEDY) async_copy_b128(baseN + (unsigned)t * 16u, g, nrow);
    }

    #pragma unroll 1
    for (int s = 0; s < STAGES; ++s) {
        const int b = s & 1;
        if (s + 1 < STAGES) {
            const int nb = b ^ 1;
            unsigned g = (segv + (unsigned)((s + 1) * THREADS + t)) * 16u;
            async_copy_b128(baseL + (unsigned)(nb * THREADS + t) * 16u, g, lrow);
            if (!GREEDY) async_copy_b128(baseN + (unsigned)(nb * THREADS + t) * 16u, g, nrow);
            // next stage's loads (1 or 2) may still be in flight; stage s is done
            if (GREEDY) wait_asynccnt_le<1>(); else wait_asynccnt_le<2>();
        } else {
            wait_asynccnt_le<0>();
        }

        const float4 lg = sL[b][t];
        const int e0 = (seg * VEC4_PER_SEG + s * THREADS + t) * 4;
        if (GREEDY) {
            upd(lg.x, e0 + 0, bv, bi);
            upd(lg.y, e0 + 1, bv, bi);
            upd(lg.z, e0 + 2, bv, bi);
            upd(lg.w, e0 + 3, bv, bi);
        } else {
            const float4 nz = sN[b][t];
            upd(fmaf(lg.x, invt, -__logf(fmaxf(nz.x, 1e-10f))), e0 + 0, bv, bi);
            upd(fmaf(lg.y, invt, -__logf(fmaxf(nz.y, 1e-10f))), e0 + 1, bv, bi);
            upd(fmaf(lg.z, invt, -__logf(fmaxf(nz.z, 1e-10f))), e0 + 2, bv, bi);
            upd(fmaf(lg.w, invt, -__logf(fmaxf(nz.w, 1e-10f))), e0 + 3, bv, bi);
        }
    }
}

__global__ __launch_bounds__(THREADS)
void sampler_partial(const float* __restrict__ logits,
                     const float* __restrict__ temps,
                     const float* __restrict__ noise,
                     float* __restrict__ pval,
                     int*   __restrict__ pidx) {
    __shared__ float4 sL[2][THREADS];   // logits staging (double buffered)
    __shared__ float4 sN[2][THREADS];   // noise staging
    __shared__ float  rv[THREADS];
    __shared__ int    ri[THREADS];

    const int t   = threadIdx.x;
    const int seg = blockIdx.x;
    const int row = blockIdx.y;

    const float tp     = temps[row];
    const bool  greedy = (tp == 0.0f);
    const float invt   = greedy ? 1.0f : (1.0f / tp);
    const float* lrow  = logits + (size_t)row * V_COLS;
    const float* nrow  = noise  + (size_t)row * V_COLS;

    float bv = -INFINITY;
    int   bi = 0x7fffffff;

    if (greedy) stream_segment<true >(lrow, nrow, invt, seg, t, sL, sN, bv, bi);
    else        stream_segment<false>(lrow, nrow, invt, seg, t, sL, sN, bv, bi);

    // block-wide argmax (LDS tree, wave32-safe)
    rv[t] = bv; ri[t] = bi;
    __syncthreads();
    #pragma unroll
    for (int off = THREADS / 2; off > 0; off >>= 1) {
        if (t < off) {
            float v2 = rv[t + off]; int i2 = ri[t + off];
            if (v2 > rv[t] || (v2 == rv[t] && i2 < ri[t])) { rv[t] = v2; ri[t] = i2; }
        }
        __syncthreads();
    }
    if (t == 0) {
        pval[row * SEGS + seg] = rv[0];
        pidx[row * SEGS + seg] = ri[0];
    }
}

__global__ __launch_bounds__(B_ROWS)
void sampler_final(const float* __restrict__ pval,
                   const int*   __restrict__ pidx,
                   float* __restrict__ out) {
    const int r = threadIdx.x;
    float bv = -INFINITY;
    int   bi = 0x7fffffff;
    #pragma unroll
    for (int s = 0; s < SEGS; ++s) {
        float v = pval[r * SEGS + s];
        int   i = pidx[r * SEGS + s];
        if (v > bv || (v == bv && i < bi)) { bv = v; bi = i; }
    }
    out[r] = (float)bi;   // argmax index (exactly representable: < 2^17)
}

extern "C" void kernel_launch(void* const* d_in, const int* in_sizes, int n_in,
                              void* d_out, int out_size, void* d_ws, size_t ws_size,
                              hipStream_t stream) {
    const float* logits = (const float*)d_in[0];   // [B, V] f32
    const float* temps  = (const float*)d_in[1];   // [B]    f32
    const float* noise  = (const float*)d_in[2];   // [B, V] f32 (Exp(1))
    float* out = (float*)d_out;                    // [B] argmax indices

    float* pv = (float*)d_ws;
    int*   pi = (int*)((char*)d_ws + (size_t)B_ROWS * SEGS * sizeof(float));

    sampler_partial<<<dim3(SEGS, B_ROWS), THREADS, 0, stream>>>(logits, temps, noise, pv, pi);
    sampler_final<<<1, B_ROWS, 0, stream>>>(pv, pi, out);
}